// RNNLayer_2723009266454
// MI455X (gfx1250) — compile-verified
//
#include <hip/hip_runtime.h>
#include <math.h>

// Problem constants (match reference).
#define T_DIM 256
#define N_DIM 256
#define DK    512          // D_IN == H == 512
#define H_DIM 512
#define H3    (3 * H_DIM)
#define LN_EPS 1e-5f
#define SCAN_BLOCKS (16 * 32)   // (N/16) x (H/16) tile waves

typedef __attribute__((ext_vector_type(2))) float v2f;
typedef __attribute__((ext_vector_type(8))) float v8f;

// Native fp32 WMMA: D(16x16,f32) = A(16x4,f32) * B(4x16,f32) + C
__device__ __forceinline__ v8f wmma_f32(v2f a, v2f b, v8f c) {
  return __builtin_amdgcn_wmma_f32_16x16x4_f32(
      /*neg_a=*/false, a, /*neg_b=*/false, b,
      /*c_mod=*/(short)0, c, /*reuse_a=*/false, /*reuse_b=*/false);
}

// Device-wide generation barrier: monotonic counter, one arrive per wave.
// Release on arrive publishes this step's h_new; acquire on the spin load
// invalidates near caches so next step's Hprev reads are coherent.
__device__ __forceinline__ void grid_barrier(unsigned* counter, unsigned target) {
  if (threadIdx.x == 0) {
    __hip_atomic_fetch_add(counter, 1u, __ATOMIC_RELEASE, __HIP_MEMORY_SCOPE_AGENT);
  }
  while (__hip_atomic_load(counter, __ATOMIC_ACQUIRE, __HIP_MEMORY_SCOPE_AGENT) < target) {
    __builtin_amdgcn_s_sleep(1);
  }
}

// ---------------------------------------------------------------------------
// Kernel 1: per-timestep reset flag: flag[t] = (t==0) || any(mask[t,:] == 1)
// Also zeroes the scan barrier counter (fresh every call -> deterministic).
// ---------------------------------------------------------------------------
__global__ void mask_flags_kernel(const float* __restrict__ masks,
                                  int* __restrict__ flags,
                                  unsigned* __restrict__ counter) {
  __shared__ int any_s;
  const int t = blockIdx.x;
  if (threadIdx.x == 0) {
    any_s = (t == 0) ? 1 : 0;
    if (t == 0) *counter = 0u;
  }
  __syncthreads();
  if (masks[t * N_DIM + threadIdx.x] == 1.0f) any_s = 1;  // benign race
  __syncthreads();
  if (threadIdx.x == 0) flags[t] = any_s;
}

// ---------------------------------------------------------------------------
// Kernel 2: GX[T*N, 3H] = X[T*N, D] @ Wih^T + bih  (hoisted out of the scan).
// Grid: (T*N/16, 3H/64), block = 128 (4 waves sharing 16 A rows for L0 reuse).
// ---------------------------------------------------------------------------
__global__ void gx_gemm_kernel(const float* __restrict__ X,
                               const float* __restrict__ Wih,
                               const float* __restrict__ bih,
                               float* __restrict__ GX) {
  const int wave = threadIdx.x >> 5;
  const int lane = threadIdx.x & 31;
  const int lm   = lane & 15;
  const int lh   = lane >> 4;
  const int row0 = blockIdx.x * 16;
  const int col0 = (blockIdx.y * 4 + wave) * 16;

  const float* Ap = X   + (size_t)(row0 + lm) * DK + 2 * lh;
  const float* Bp = Wih + (size_t)(col0 + lm) * DK + 2 * lh;

  v8f acc = {};
#pragma unroll 4
  for (int k = 0; k < DK; k += 4) {
    v2f a = *(const v2f*)(Ap + k);
    v2f b = *(const v2f*)(Bp + k);
    acc = wmma_f32(a, b, acc);
  }

  const float bias = bih[col0 + lm];
  float* Cp = GX + (size_t)(row0 + 8 * lh) * H3 + col0 + lm;
#pragma unroll
  for (int v = 0; v < 8; ++v) Cp[(size_t)v * H3] = acc[v] + bias;
}

// ---------------------------------------------------------------------------
// Kernel 3: persistent GRU scan. 512 tile-waves stay resident; each owns one
// 16x16 tile of h and loops over all T steps, separated by a device-wide
// barrier. W_hh addressing and gate biases are hoisted out of the time loop.
// ys[t] is written into d_out and doubles as Hprev for step t+1.
// ---------------------------------------------------------------------------
__global__ void gru_scan_kernel(const float* __restrict__ GX,    // [T*N, 3H]
                                const float* __restrict__ H0,    // [N, H]
                                const float* __restrict__ Whh,   // [3H, H]
                                const float* __restrict__ bhh,   // [3H]
                                const float* __restrict__ masks, // [T*N]
                                const int* __restrict__ flags,   // [T]
                                float* __restrict__ out,         // [T*N, H]
                                unsigned* __restrict__ counter) {
  const int lane = threadIdx.x & 31;
  const int lm   = lane & 15;
  const int lh   = lane >> 4;
  const int row0 = blockIdx.x * 16;   // batch rows
  const int col0 = blockIdx.y * 16;   // hidden cols
  const int col  = col0 + lm;

  // Time-invariant: weight fragment pointers and gate biases.
  const float* Brp = Whh + (size_t)(col0 + lm) * H_DIM + 2 * lh;
  const float* Bzp = Brp + (size_t)H_DIM * H_DIM;
  const float* Bnp = Bzp + (size_t)H_DIM * H_DIM;
  const float bhr = bhh[col];
  const float bhz = bhh[H_DIM + col];
  const float bhn = bhh[2 * H_DIM + col];

  for (int t = 0; t < T_DIM; ++t) {
    const float* Hprev = (t == 0) ? H0 : out + (size_t)(t - 1) * N_DIM * H_DIM;
    const float* GXt   = GX + (size_t)t * N_DIM * H3;
    const float* Mt    = masks + (size_t)t * N_DIM;
    float*       Hout  = out + (size_t)t * N_DIM * H_DIM;

    const int   flag = flags[t];
    const float sA   = flag ? Mt[row0 + lm] : 1.0f;

    const float* Ap = Hprev + (size_t)(row0 + lm) * H_DIM + 2 * lh;

    v8f ar = {}, az = {}, an = {};
#pragma unroll 2
    for (int k = 0; k < H_DIM; k += 4) {
      v2f a = *(const v2f*)(Ap + k);
      a.x *= sA; a.y *= sA;                 // masked hidden state feeds GEMM
      v2f br = *(const v2f*)(Brp + k);
      v2f bz = *(const v2f*)(Bzp + k);
      v2f bn = *(const v2f*)(Bnp + k);
      ar = wmma_f32(a, br, ar);
      az = wmma_f32(a, bz, az);
      an = wmma_f32(a, bn, an);
    }

#pragma unroll
    for (int v = 0; v < 8; ++v) {
      const int    row = row0 + v + 8 * lh;
      const size_t gxb = (size_t)row * H3;
      const float ghr = ar[v] + bhr;
      const float ghz = az[v] + bhz;
      const float ghn = an[v] + bhn;
      const float r = 1.0f / (1.0f + expf(-(GXt[gxb + col] + ghr)));
      const float z = 1.0f / (1.0f + expf(-(GXt[gxb + H_DIM + col] + ghz)));
      const float n = tanhf(GXt[gxb + 2 * H_DIM + col] + r * ghn);
      const float hp   = Hprev[(size_t)row * H_DIM + col];
      const float heff = flag ? hp * Mt[row] : hp;
      Hout[(size_t)row * H_DIM + col] = (1.0f - z) * n + z * heff;
    }

    if (t + 1 < T_DIM) {
      // Prefetch next step's gx tile (streams from HBM; hide latency behind
      // the barrier). Speculative global_prefetch_b8, one line per row/gate.
      const char* pf = (const char*)(GXt + (size_t)N_DIM * H3 +
                                     (size_t)(row0 + lm) * H3 + col0 + lh * H_DIM);
      __builtin_prefetch(pf, 0, 0);                              // r / z gates
      if (lh == 0) __builtin_prefetch(pf + 2 * H_DIM * 4, 0, 0); // n gate
      grid_barrier(counter, (unsigned)(t + 1) * SCAN_BLOCKS);
    }
  }
}

// ---------------------------------------------------------------------------
// Kernel 4: row LayerNorm in place over out[T*N, H]; copies the raw (pre-LN)
// last-step rows into hxs_out first.
// ---------------------------------------------------------------------------
__global__ void layernorm_kernel(float* __restrict__ out,
                                 const float* __restrict__ gamma,
                                 const float* __restrict__ beta,
                                 float* __restrict__ hT_out) {
  const int row = blockIdx.x;
  const int tid = threadIdx.x;
  float* rp = out + (size_t)row * H_DIM;
  const float x0 = rp[tid];
  const float x1 = rp[tid + 256];

  if (row >= (T_DIM - 1) * N_DIM) {   // hT = ys[T-1] (pre-LayerNorm)
    float* hp = hT_out + (size_t)(row - (T_DIM - 1) * N_DIM) * H_DIM;
    hp[tid] = x0;
    hp[tid + 256] = x1;
  }

  float sum = x0 + x1;
  float sq  = x0 * x0 + x1 * x1;
#pragma unroll
  for (int off = 16; off > 0; off >>= 1) {   // wave32 reduction
    sum += __shfl_down(sum, off, 32);
    sq  += __shfl_down(sq, off, 32);
  }
  __shared__ float s_sum[8], s_sq[8], s_mu, s_rstd;
  const int wave = tid >> 5;
  if ((tid & 31) == 0) { s_sum[wave] = sum; s_sq[wave] = sq; }
  __syncthreads();
  if (tid == 0) {
    float ts = 0.f, tq = 0.f;
    for (int i = 0; i < 8; ++i) { ts += s_sum[i]; tq += s_sq[i]; }
    const float mu  = ts * (1.0f / H_DIM);
    const float var = tq * (1.0f / H_DIM) - mu * mu;
    s_mu = mu; s_rstd = rsqrtf(var + LN_EPS);
  }
  __syncthreads();
  const float mu = s_mu, rstd = s_rstd;
  rp[tid]       = (x0 - mu) * rstd * gamma[tid]       + beta[tid];
  rp[tid + 256] = (x1 - mu) * rstd * gamma[tid + 256] + beta[tid + 256];
}

// ---------------------------------------------------------------------------
extern "C" void kernel_launch(void* const* d_in, const int* in_sizes, int n_in,
                              void* d_out, int out_size, void* d_ws, size_t ws_size,
                              hipStream_t stream) {
  (void)in_sizes; (void)n_in; (void)out_size; (void)ws_size;
  const float* x     = (const float*)d_in[0];
  const float* hxs   = (const float*)d_in[1];  // [N,1,H] -> flat [N,H]
  const float* masks = (const float*)d_in[2];  // [T*N,1]
  const float* wih   = (const float*)d_in[3];  // [3H, D]
  const float* whh   = (const float*)d_in[4];  // [3H, H]
  const float* bih   = (const float*)d_in[5];
  const float* bhh   = (const float*)d_in[6];
  const float* gamma = (const float*)d_in[7];
  const float* beta  = (const float*)d_in[8];

  float* out = (float*)d_out;                                  // [T*N, H]
  float* hT  = out + (size_t)T_DIM * N_DIM * H_DIM;            // [N, H]

  // Workspace: GX [T*N, 3H] f32 (~384 MB) + per-t flags + barrier counter.
  float*    gx      = (float*)d_ws;
  char*     wsrest  = (char*)d_ws + (size_t)T_DIM * N_DIM * H3 * sizeof(float);
  int*      flags   = (int*)wsrest;
  unsigned* counter = (unsigned*)(wsrest + T_DIM * sizeof(int));

  mask_flags_kernel<<<T_DIM, N_DIM, 0, stream>>>(masks, flags, counter);

  gx_gemm_kernel<<<dim3((T_DIM * N_DIM) / 16, H3 / 64), 128, 0, stream>>>(
      x, wih, bih, gx);

  // Persistent scan: all 256 timesteps in one launch, device-wide barrier
  // between steps (512 single-wave blocks are always co-resident).
  gru_scan_kernel<<<dim3(N_DIM / 16, H_DIM / 16), 32, 0, stream>>>(
      gx, hxs, whh, bhh, masks, flags, out, counter);

  layernorm_kernel<<<T_DIM * N_DIM, 256, 0, stream>>>(out, gamma, beta, hT);
}